// NeuroGraphTokenizer_47596827574734
// MI455X (gfx1250) — compile-verified
//
#include <hip/hip_runtime.h>

typedef __attribute__((ext_vector_type(2))) float v2f;
typedef __attribute__((ext_vector_type(8))) float v8f;

#define TPB 256

// ---------------------------------------------------------------------------
// Degree / normalization (shared by all 5 gcn_conv layers)
// ---------------------------------------------------------------------------
__global__ void k_deg_init(float* __restrict__ deg, int N) {
  int i = blockIdx.x * blockDim.x + threadIdx.x;
  if (i < N) deg[i] = 1.0f;  // self-loop contributes 1 to every node's degree
}

__global__ void k_deg_edges(const long long* __restrict__ dst,
                            float* __restrict__ deg, int E) {
  int e = blockIdx.x * blockDim.x + threadIdx.x;
  if (e < E) atomicAdd(&deg[(int)dst[e]], 1.0f);
}

__global__ void k_deg_to_dinv(float* __restrict__ deg, int N) {
  int i = blockIdx.x * blockDim.x + threadIdx.x;
  if (i < N) deg[i] = rsqrtf(deg[i]);  // deg >= 1 always (self loops)
}

// ---------------------------------------------------------------------------
// Pack W[Fin,Fout] into WMMA B-fragment order:
//   Wp[(g*Fout + c)*4 + i] = W[(4g+i)*Fout + c],  g = k-group, i = row-in-group
// so that lane (n = l&15, kh = l>>4) of a wave reads its (B[k+2kh][n],
// B[k+2kh+1][n]) pair as ONE b64 load, coalesced to 256 contiguous bytes
// per fragment across the wave.
// ---------------------------------------------------------------------------
template <int Fin, int Fout>
__global__ void k_pack_w(const float* __restrict__ W, float* __restrict__ Wp) {
  int t = blockIdx.x * blockDim.x + threadIdx.x;
  if (t >= Fin * Fout) return;
  int r = t / Fout;                  // compile-time Fout -> shift
  int c = t - r * Fout;
  int g = r >> 2;
  int i = r & 3;
  Wp[(((size_t)g * Fout + c) << 2) + i] = W[t];
}

// ---------------------------------------------------------------------------
// f32 WMMA GEMM: H[N,Fout] = X[N,Fin] @ W[Fin,Fout], W pre-packed as above.
// One wave computes a 16x64 tile (4 accumulators reuse each A fragment 4x),
// K stepped by 4 via V_WMMA_F32_16X16X4_F32 (exact f32, matches reference).
// All strides compile-time -> loads use immediate offsets, no 64-bit VALU
// address chains in the inner loop.
//
// A frag (16x4 f32, ISA 7.12.2): lane m=l&15, kh=l>>4: v0=A[m][k+2kh],
//                                v1=A[m][k+1+2kh]
// C/D (16x16 f32): vgpr r, lanes 0-15 -> M=r, lanes 16-31 -> M=r+8, N=l&15
// ---------------------------------------------------------------------------
template <int Fin, int Fout>
__global__ void k_gemm_wmma_f32(const float* __restrict__ X,
                                const float* __restrict__ Wp,
                                float* __restrict__ H, int N) {
  constexpr int NTC = Fout >> 6;           // 64-wide column tiles
  int wave = (int)((blockIdx.x * (long long)blockDim.x + threadIdx.x) >> 5);
  int lane = threadIdx.x & 31;
  int rt   = wave / NTC;                   // constexpr divisor -> shifts
  int ct   = wave - rt * NTC;
  int row0 = rt << 4;
  if (row0 + 16 > N) return;
  int col0 = ct << 6;

  int m   = lane & 15;
  int kh2 = (lane >> 4) << 1;              // 0 or 2 (k-half * 2)

  v8f acc0 = {}, acc1 = {}, acc2 = {}, acc3 = {};

  const float* xp = X  + (size_t)(row0 + m) * Fin + kh2;
  const float* wp = Wp + (((size_t)col0 + m) << 2) + kh2;

#pragma unroll 16
  for (int g = 0; g < Fin / 4; ++g) {
    v2f a  = *(const v2f*)(xp + 4 * g);
    v2f b0 = *(const v2f*)(wp + (size_t)g * 4 * Fout);
    v2f b1 = *(const v2f*)(wp + (size_t)g * 4 * Fout + 64);
    v2f b2 = *(const v2f*)(wp + (size_t)g * 4 * Fout + 128);
    v2f b3 = *(const v2f*)(wp + (size_t)g * 4 * Fout + 192);
    acc0 = __builtin_amdgcn_wmma_f32_16x16x4_f32(false, a, false, b0, (short)0, acc0, false, false);
    acc1 = __builtin_amdgcn_wmma_f32_16x16x4_f32(false, a, false, b1, (short)0, acc1, false, false);
    acc2 = __builtin_amdgcn_wmma_f32_16x16x4_f32(false, a, false, b2, (short)0, acc2, false, false);
    acc3 = __builtin_amdgcn_wmma_f32_16x16x4_f32(false, a, false, b3, (short)0, acc3, false, false);
  }

  int crow = row0 + ((lane >> 4) << 3);    // +8 for upper half-lanes
  float* hp = H + (size_t)crow * Fout + col0 + m;
#pragma unroll
  for (int r = 0; r < 8; ++r) {
    hp[(size_t)r * Fout +  0] = acc0[r];
    hp[(size_t)r * Fout + 16] = acc1[r];
    hp[(size_t)r * Fout + 32] = acc2[r];
    hp[(size_t)r * Fout + 48] = acc3[r];
  }
}

// ---------------------------------------------------------------------------
// Aggregation: out = segsum_{dst}(H[src]*dinv[src]*dinv[dst]) + H*dinv^2 + b
// Split: init (self-loop term + bias, full overwrite), then edge atomics.
// All layer working sets are L2-resident (192MB L2), so the f32 atomics
// resolve at L2 rates, not HBM.
// ---------------------------------------------------------------------------
__global__ void k_agg_init(const float* __restrict__ H,
                           const float* __restrict__ dinv,
                           const float* __restrict__ bias,
                           float* __restrict__ out,
                           long long total, int lf) {
  long long t = blockIdx.x * (long long)blockDim.x + threadIdx.x;
  if (t >= total) return;
  int i = (int)(t >> lf);
  int f = (int)(t & ((1 << lf) - 1));
  float di = dinv[i];
  out[t] = H[t] * di * di + bias[f];
}

__global__ void k_agg_edges(const float* __restrict__ H,
                            const float* __restrict__ dinv,
                            const long long* __restrict__ src,
                            const long long* __restrict__ dst,
                            float* __restrict__ out,
                            long long total, int lf, int lc) {
  long long t = blockIdx.x * (long long)blockDim.x + threadIdx.x;
  if (t >= total) return;
  int e = (int)(t >> lc);
  int c = (int)(t & ((1 << lc) - 1));
  int s = (int)src[e];
  int d = (int)dst[e];
  float nrm = dinv[s] * dinv[d];
  const float4 h = *((const float4*)(H + ((size_t)s << lf)) + c);
  float* op = out + ((size_t)d << lf) + ((size_t)c << 2);
  atomicAdd(op + 0, h.x * nrm);
  atomicAdd(op + 1, h.y * nrm);
  atomicAdd(op + 2, h.z * nrm);
  atomicAdd(op + 3, h.w * nrm);
}

__global__ void k_relu(float* __restrict__ out, long long n) {
  long long t = blockIdx.x * (long long)blockDim.x + threadIdx.x;
  if (t < n) out[t] = fmaxf(out[t], 0.0f);
}

// ---------------------------------------------------------------------------
static inline unsigned nblk(long long n) { return (unsigned)((n + TPB - 1) / TPB); }

template <int Fin, int Fout>
static void gcn_layer(const float* X, const float* W, const float* b,
                      float* Wp, float* Hbuf, float* out,
                      const long long* src, const long long* dst,
                      const float* dinv,
                      int N, int E, bool relu, hipStream_t stream) {
  k_pack_w<Fin, Fout><<<nblk(Fin * Fout), TPB, 0, stream>>>(W, Wp);

  long long thr = (long long)(N / 16) * (Fout / 64) * 32;
  k_gemm_wmma_f32<Fin, Fout><<<nblk(thr), TPB, 0, stream>>>(X, Wp, Hbuf, N);

  int lf = 31 - __builtin_clz((unsigned)Fout);   // Fout = 128/256/512
  long long nn = (long long)N << lf;
  k_agg_init<<<nblk(nn), TPB, 0, stream>>>(Hbuf, dinv, b, out, nn, lf);

  int lc = lf - 2;                               // float4 chunks per row
  long long ne = (long long)E << lc;
  k_agg_edges<<<nblk(ne), TPB, 0, stream>>>(Hbuf, dinv, src, dst, out, ne, lf, lc);

  if (relu) k_relu<<<nblk(nn), TPB, 0, stream>>>(out, nn);
}

extern "C" void kernel_launch(void* const* d_in, const int* in_sizes, int n_in,
                              void* d_out, int out_size, void* d_ws, size_t ws_size,
                              hipStream_t stream) {
  const float*     v  = (const float*)d_in[0];
  const long long* ei = (const long long*)d_in[1];
  const float* W1 = (const float*)d_in[2];  const float* b1 = (const float*)d_in[3];
  const float* W2 = (const float*)d_in[4];  const float* b2 = (const float*)d_in[5];
  const float* W3 = (const float*)d_in[6];  const float* b3 = (const float*)d_in[7];
  const float* Wm = (const float*)d_in[8];  const float* bm = (const float*)d_in[9];
  const float* Ws = (const float*)d_in[10]; const float* bs = (const float*)d_in[11];

  const int N = in_sizes[0] / 64;   // 50000
  const int E = in_sizes[1] / 2;    // 800000
  const long long* src = ei;
  const long long* dst = ei + E;

  // Workspace: dinv[N] | bufX[N*512] | bufH45[N*256] | Wp[512*512]  (~155 MB)
  float* ws     = (float*)d_ws;
  float* dinv   = ws;
  float* bufX   = ws + (((size_t)N + 255) & ~(size_t)255);
  float* bufH45 = bufX + (size_t)N * 512;
  float* Wp     = bufH45 + (size_t)N * 256;

  // d_out (= N*512 floats total) doubles as H scratch for layers 1-3.
  float* out_mu  = (float*)d_out;
  float* out_std = out_mu + (size_t)N * 256;
  float* bufH123 = (float*)d_out;

  // --- normalization (computed once, reused for all layers) ---
  k_deg_init   <<<nblk(N), TPB, 0, stream>>>(dinv, N);
  k_deg_edges  <<<nblk(E), TPB, 0, stream>>>(dst, dinv, E);
  k_deg_to_dinv<<<nblk(N), TPB, 0, stream>>>(dinv, N);

  // --- 5 GCN layers ---
  gcn_layer< 64, 128>(v,    W1, b1, Wp, bufH123, bufX,    src, dst, dinv, N, E, true,  stream);
  gcn_layer<128, 256>(bufX, W2, b2, Wp, bufH123, bufX,    src, dst, dinv, N, E, true,  stream);
  gcn_layer<256, 512>(bufX, W3, b3, Wp, bufH123, bufX,    src, dst, dinv, N, E, true,  stream);
  // heads: H3 (in d_out) is dead now; mu/std H goes to bufH45, results to d_out
  gcn_layer<512, 256>(bufX, Wm, bm, Wp, bufH45,  out_mu,  src, dst, dinv, N, E, false, stream);
  gcn_layer<512, 256>(bufX, Ws, bs, Wp, bufH45,  out_std, src, dst, dinv, N, E, false, stream);
}